// GraphSumEdgeConv_63170378989709
// MI455X (gfx1250) — compile-verified
//
#include <hip/hip_runtime.h>

typedef float v2f __attribute__((ext_vector_type(2)));
typedef float v8f __attribute__((ext_vector_type(8)));

#define DFEAT 128
#define KHALF 64
#define LDSW  68   // 64 + 4 pad: LDS word bank = (4n + k) % 64 -> conflict-free b64 reads

// Fused GEMM (Y' = Y * W^T, fp32 WMMA 16x16x4) + scatter-add into agg.
// One wave computes rows [rb*16, rb*16+16) of Y' across all 128 output features.
__global__ __launch_bounds__(256) void gsec_gemm_scatter(
    const float* __restrict__ Y,
    const float* __restrict__ W,
    const long long* __restrict__ edge_index,
    float* __restrict__ Yp,
    float* __restrict__ agg,
    int E)
{
    __shared__ float sW[DFEAT * LDSW];   // W[:, phase*64 .. +64), row-major, padded

    const int lane    = threadIdx.x & 31;
    const int wave    = threadIdx.x >> 5;
    const int rb      = blockIdx.x * 8 + wave;   // 16-row block id
    const int r0      = rb * 16;
    const int halfSel = lane >> 4;               // 0: lanes 0-15, 1: lanes 16-31
    const int nl      = lane & 15;
    const int koff    = halfSel * 2;             // K sub-offset per ISA A/B layout

    v8f acc[8] = {};

    // A row for this lane (clamped so tail blocks don't read OOB)
    int rA = r0 + nl;
    if (rA >= E) rA = (E > 0) ? (E - 1) : 0;
    const float* __restrict__ Arow = Y + (size_t)rA * DFEAT;

    for (int phase = 0; phase < 2; ++phase) {
        __syncthreads();
        // Cooperative load of one 128x64 half of W into LDS (float4 per thread-step)
        for (int idx = threadIdx.x; idx < DFEAT * (KHALF / 4); idx += 256) {
            const int n  = idx >> 4;          // 16 float4 per row
            const int k4 = (idx & 15) << 2;
            const float4 w = *(const float4*)(W + (size_t)n * DFEAT + phase * KHALF + k4);
            float* dst = &sW[n * LDSW + k4];
            dst[0] = w.x; dst[1] = w.y; dst[2] = w.z; dst[3] = w.w;
        }
        __syncthreads();

        if (r0 < E) {
            #pragma unroll 4
            for (int kk = 0; kk < KHALF; kk += 4) {
                const int k0 = phase * KHALF + kk;
                // A fragment: 16x4 fp32 tile of Y
                const v2f a = *(const v2f*)(Arow + k0 + koff);
                #pragma unroll
                for (int nt = 0; nt < 8; ++nt) {
                    // B fragment: 4x16 tile of W^T, i.e. B[k][n] = W[n][k]
                    const v2f b = *(const v2f*)(&sW[(nt * 16 + nl) * LDSW + kk + koff]);
                    acc[nt] = __builtin_amdgcn_wmma_f32_16x16x4_f32(
                        false, a, false, b, (short)0, acc[nt], false, false);
                }
            }
        }
    }

    if (r0 >= E) return;

    // Write Y' tile and scatter-add into agg[src[e], :]
    #pragma unroll
    for (int v = 0; v < 8; ++v) {
        const int gr = r0 + v + halfSel * 8;   // C/D layout: lanes 0-15 -> M=v, 16-31 -> M=v+8
        if (gr < E) {
            const int s = (int)edge_index[(size_t)gr * 2];  // src = edge_index[e, 0]
            float* __restrict__ yprow  = Yp  + (size_t)gr * DFEAT + nl;
            float* __restrict__ aggrow = agg + (size_t)s  * DFEAT + nl;
            #pragma unroll
            for (int nt = 0; nt < 8; ++nt) {
                const float val = acc[nt][v];
                yprow[nt * 16] = val;
                __hip_atomic_fetch_add(&aggrow[nt * 16], val,
                                       __ATOMIC_RELAXED, __HIP_MEMORY_SCOPE_AGENT);
            }
        }
    }
}

__global__ void gsec_zero(float4* __restrict__ p, int n4)
{
    const int i = blockIdx.x * blockDim.x + threadIdx.x;
    if (i < n4) p[i] = make_float4(0.f, 0.f, 0.f, 0.f);
}

__global__ void gsec_relu_add(const float4* __restrict__ X,
                              const float4* __restrict__ agg,
                              float4* __restrict__ Xp, int n4)
{
    const int i = blockIdx.x * blockDim.x + threadIdx.x;
    if (i < n4) {
        const float4 x = X[i];
        const float4 a = agg[i];
        float4 r;
        r.x = fmaxf(x.x + a.x, 0.f);
        r.y = fmaxf(x.y + a.y, 0.f);
        r.z = fmaxf(x.z + a.z, 0.f);
        r.w = fmaxf(x.w + a.w, 0.f);
        Xp[i] = r;
    }
}

extern "C" void kernel_launch(void* const* d_in, const int* in_sizes, int n_in,
                              void* d_out, int out_size, void* d_ws, size_t ws_size,
                              hipStream_t stream)
{
    const float*     X  = (const float*)d_in[0];
    const float*     Y  = (const float*)d_in[1];
    const float*     W  = (const float*)d_in[2];
    const long long* EI = (const long long*)d_in[3];

    const int N = in_sizes[0] / DFEAT;   // 10000 nodes
    const int E = in_sizes[1] / DFEAT;   // 640000 edges

    float* Xp  = (float*)d_out;                        // N*128 floats
    float* Yp  = (float*)d_out + (size_t)N * DFEAT;    // E*128 floats
    float* agg = (float*)d_ws;                         // N*128 floats scratch

    // 1) zero the aggregation buffer
    const int n4 = (N * DFEAT) / 4;
    gsec_zero<<<(n4 + 255) / 256, 256, 0, stream>>>((float4*)agg, n4);

    // 2) fused WMMA GEMM + scatter-add (8 waves/block, 16 edge-rows per wave)
    const int blocks = (E + 127) / 128;
    gsec_gemm_scatter<<<blocks, 256, 0, stream>>>(Y, W, EI, Yp, agg, E);

    // 3) X' = relu(X + agg)
    gsec_relu_add<<<(n4 + 255) / 256, 256, 0, stream>>>(
        (const float4*)X, (const float4*)agg, (float4*)Xp, n4);
}